// Fusionloss_LS2_32908039422557
// MI455X (gfx1250) — compile-verified
//
#include <hip/hip_runtime.h>
#include <hip/hip_bf16.h>
#include <stdint.h>

// ---------------------------------------------------------------------------
// Fusionloss on MI455X (gfx1250).
// Memory-bound stencil+reduction: ~20MB live data, L2-resident, ideal ~1us.
// Strategy: single-pass fused tiles, async global->LDS staging (CDNA5 async
// path, ASYNCcnt-tracked), wave32 shuffle reductions, deterministic two-stage
// reductions (no float atomics).
// ---------------------------------------------------------------------------

#define B_   4
#define H_   640
#define W_   640
#define NL_  10
#define TS   32          // tile size (640 % 32 == 0 -> all tiles full)
#define HS   34          // tile + 1-halo
#define LST  35          // padded LDS row stride (odd -> fewer bank conflicts)
#define NTX  (W_/TS)     // 20
#define NTY  (H_/TS)     // 20
#define NT   (NTX*NTY)   // 400 tiles per image
#define NPIX ((float)(H_*W_))
#define NWAVES 8         // 256 threads / wave32

// Low 32 bits of a flat pointer to LDS == LDS byte offset (aperture rule).
__device__ __forceinline__ uint32_t lds_off32(const void* p) {
  return (uint32_t)(uintptr_t)p;
}

// CDNA5 async global->LDS copy, per-lane LDS destination address.
__device__ __forceinline__ void async_g2l_b32(uint32_t lds_byte, const float* g) {
  asm volatile("global_load_async_to_lds_b32 %0, %1, off"
               :: "v"(lds_byte), "v"(g) : "memory");
}
__device__ __forceinline__ void wait_async0() {
  asm volatile("s_wait_asynccnt 0" ::: "memory");
}

// 3x3 Sobel (cross-correlation, matches lax/torch conv) around center idx ci.
__device__ __forceinline__ void sobel_xy(const float* sh, int ci, float& gx, float& gy) {
  float nw = sh[ci - LST - 1], nn = sh[ci - LST], ne = sh[ci - LST + 1];
  float ww = sh[ci - 1],                           ee = sh[ci + 1];
  float sw = sh[ci + LST - 1], ss = sh[ci + LST], se = sh[ci + LST + 1];
  gx = (ne - nw) + 2.0f * (ee - ww) + (se - sw);
  gy = (nw + 2.0f * nn + ne) - (sw + 2.0f * ss + se);
}

// Deterministic 256-thread dual reduction: wave32 xor-butterfly (no LDS, no
// barriers) + fixed-order 8-wave LDS combine. Valid result on tid==0 only.
__device__ __forceinline__ float2 block_reduce2(float a, float b,
                                                float* redA, float* redB,
                                                int tid) {
  #pragma unroll
  for (int off = 16; off > 0; off >>= 1) {
    a += __shfl_xor(a, off, 32);
    b += __shfl_xor(b, off, 32);
  }
  const int wave = tid >> 5, lane = tid & 31;
  if (lane == 0) { redA[wave] = a; redB[wave] = b; }
  __syncthreads();
  float2 r = make_float2(0.0f, 0.0f);
  if (tid == 0) {
    float sa = redA[0], sb = redB[0];
    #pragma unroll
    for (int w = 1; w < NWAVES; ++w) { sa += redA[w]; sb += redB[w]; }
    r = make_float2(sa, sb);
  }
  __syncthreads();   // protect redA/redB for reuse by caller
  return r;
}

// -------------------- Kernel A: global losses, tile partials ----------------
__global__ __launch_bounds__(256)
void fusion_global_tiles(const float* __restrict__ vis,
                         const float* __restrict__ th,
                         const float* __restrict__ gen,
                         float* __restrict__ part)   // [B][NT][2]
{
  __shared__ float sv[HS * LST], st[HS * LST], sg[HS * LST];
  __shared__ float redA[NWAVES], redB[NWAVES];

  const int bi  = blockIdx.z;
  const int r0  = blockIdx.y * TS;
  const int c0  = blockIdx.x * TS;
  const int tid = threadIdx.y * 32 + threadIdx.x;
  const size_t base = (size_t)bi * (size_t)(H_ * W_);

  // Stage tile+halo. In-image lanes: async global->LDS. Out-of-image halo:
  // DS zero stores (disjoint LDS addresses -> no DS/async ordering hazard).
  for (int idx = tid; idx < HS * HS; idx += 256) {
    int lr = idx / HS, lc = idx % HS;
    int gr = r0 + lr - 1, gc = c0 + lc - 1;
    int li = lr * LST + lc;
    if (gr >= 0 && gr < H_ && gc >= 0 && gc < W_) {
      size_t o = base + (size_t)gr * W_ + gc;
      async_g2l_b32(lds_off32(&sv[li]), vis + o);
      async_g2l_b32(lds_off32(&st[li]), th  + o);
      async_g2l_b32(lds_off32(&sg[li]), gen + o);
    } else {
      sv[li] = 0.0f; st[li] = 0.0f; sg[li] = 0.0f;
    }
  }
  wait_async0();     // my async LDS writes landed
  __syncthreads();   // everyone's writes visible (also fences DScnt)

  float s_in = 0.0f, s_gr = 0.0f;
  #pragma unroll
  for (int k = 0; k < 4; ++k) {
    int lr = threadIdx.y + 8 * k;          // 0..31 within tile
    int lc = threadIdx.x;                  // 0..31
    int ci = (lr + 1) * LST + (lc + 1);

    float gxv, gyv, gxt, gyt, gxg, gyg;
    sobel_xy(sv, ci, gxv, gyv);
    sobel_xy(st, ci, gxt, gyt);
    sobel_xy(sg, ci, gxg, gyg);

    float y_grad   = fabsf(gxv) + fabsf(gyv);
    // sobel(0.5*vis + 0.5*th) by linearity of the per-direction conv
    float m_grad   = fabsf(0.5f * (gxv + gxt)) + fabsf(0.5f * (gyv + gyt));
    float gen_grad = fabsf(gxg) + fabsf(gyg);
    float joint    = fmaxf(y_grad, m_grad);
    s_gr += fabsf(gen_grad - joint);

    float vc = sv[ci], tc = st[ci], gc_ = sg[ci];
    float mean  = 0.5f * vc + 0.5f * tc;
    float inmax = fmaxf(vc, mean);
    s_in += fabsf(gc_ - inmax);
  }

  float2 r = block_reduce2(s_in, s_gr, redA, redB, tid);
  if (tid == 0) {
    const int slot = (bi * NT + blockIdx.y * NTX + blockIdx.x) * 2;
    part[slot + 0] = r.x;
    part[slot + 1] = r.y;
  }
}

// -------------------- Kernel B: per-box masked losses -----------------------
__global__ __launch_bounds__(256)
void fusion_box_tiles(const float* __restrict__ vis,
                      const float* __restrict__ th,
                      const float* __restrict__ gen,
                      const int*   __restrict__ label,   // [B][NL][5]
                      float* __restrict__ part)          // [B*NL][NT][2]
{
  __shared__ float sv[HS * LST], st[HS * LST], sg[HS * LST];
  __shared__ float redA[NWAVES], redB[NWAVES];

  const int box = blockIdx.z;             // bi*NL + li
  const int bi  = box / NL_;
  const int r0  = blockIdx.y * TS;
  const int c0  = blockIdx.x * TS;
  const int tid = threadIdx.y * 32 + threadIdx.x;
  const int slot = (box * NT + blockIdx.y * NTX + blockIdx.x) * 2;

  const int x1 = label[box * 5 + 1];
  const int y1 = label[box * 5 + 2];
  const int x2 = label[box * 5 + 3];
  const int y2 = label[box * 5 + 4];

  // Box does not overlap this tile -> deterministic zero partial.
  if (!(y1 < r0 + TS && y2 > r0 && x1 < c0 + TS && x2 > c0)) {
    if (tid == 0) { part[slot + 0] = 0.0f; part[slot + 1] = 0.0f; }
    return;
  }

  const size_t base = (size_t)bi * (size_t)(H_ * W_);

  // Mask applied at fill time: LDS holds the *boxed* images, so the Sobel of
  // the masked image (incl. its zeroed boundary) is exact.
  for (int idx = tid; idx < HS * HS; idx += 256) {
    int lr = idx / HS, lc = idx % HS;
    int gr = r0 + lr - 1, gc = c0 + lc - 1;
    int li = lr * LST + lc;
    bool in = (gr >= y1 && gr < y2 && gc >= x1 && gc < x2 &&
               gr >= 0 && gr < H_ && gc >= 0 && gc < W_);
    if (in) {
      size_t o = base + (size_t)gr * W_ + gc;
      sv[li] = vis[o]; st[li] = th[o]; sg[li] = gen[o];
    } else {
      sv[li] = 0.0f; st[li] = 0.0f; sg[li] = 0.0f;
    }
  }
  __syncthreads();

  float s0 = 0.0f, s1 = 0.0f;   // Lssim numerator, L1 numerator
  #pragma unroll
  for (int k = 0; k < 4; ++k) {
    int lr = threadIdx.y + 8 * k;
    int lc = threadIdx.x;
    int gr = r0 + lr, gc = c0 + lc;
    if (gr >= y1 && gr < y2 && gc >= x1 && gc < x2) {  // fmask == 1
      int ci = (lr + 1) * LST + (lc + 1);
      float gxv, gyv, gxt, gyt, gxg, gyg;
      sobel_xy(sv, ci, gxv, gyv);
      sobel_xy(st, ci, gxt, gyt);
      sobel_xy(sg, ci, gxg, gyg);

      float g_vis  = fabsf(gxv) + fabsf(gyv);
      // sobel(0.3*m_vis + 0.7*m_th) by linearity
      float g_mean = fabsf(0.3f * gxv + 0.7f * gxt) + fabsf(0.3f * gyv + 0.7f * gyt);
      float g_gen  = fabsf(gxg) + fabsf(gyg);
      s0 += fabsf(g_gen - fmaxf(g_vis, g_mean));

      float vc = sv[ci], tc = st[ci], gcv = sg[ci];
      float mean7 = 0.3f * vc + 0.7f * tc;
      s1 += fabsf(gcv - fmaxf(vc, mean7));
    }
  }

  float2 r = block_reduce2(s0, s1, redA, redB, tid);
  if (tid == 0) {
    part[slot + 0] = r.x;
    part[slot + 1] = r.y;
  }
}

// -------------------- Finalize: global losses -------------------------------
__global__ __launch_bounds__(256)
void finalize_global(const float* __restrict__ part,   // [B][NT][2]
                     const float* __restrict__ b_scalar,
                     float* __restrict__ out)           // 7 x [B]
{
  __shared__ float redA[NWAVES], redB[NWAVES];
  const int bi = blockIdx.x, tid = threadIdx.x;
  float s0 = 0.0f, s1 = 0.0f;
  for (int t = tid; t < NT; t += 256) {
    s0 += part[(bi * NT + t) * 2 + 0];
    s1 += part[(bi * NT + t) * 2 + 1];
  }
  float2 r = block_reduce2(s0, s1, redA, redB, tid);
  if (tid == 0) {
    float loss_in   = r.x / NPIX;
    float loss_grad = r.y / NPIX;
    float alpha = b_scalar[0];
    out[0 * B_ + bi] = 0.0f;                                        // loss_ss
    out[1 * B_ + bi] = alpha * loss_in + (1.0f - alpha) * loss_grad;// loss_global
    out[3 * B_ + bi] = loss_in;                                     // loss_in
    out[4 * B_ + bi] = loss_grad;                                   // loss_grad
  }
}

// -------------------- Finalize: label losses --------------------------------
__global__ __launch_bounds__(256)
void finalize_label(const float* __restrict__ part,    // [B*NL][NT][2]
                    const int*   __restrict__ label,
                    const float* __restrict__ c_scalar,
                    float* __restrict__ out)
{
  __shared__ float redA[NWAVES], redB[NWAVES];
  __shared__ float Ls[NL_], L1[NL_];
  const int bi = blockIdx.x, tid = threadIdx.x;

  for (int li = 0; li < NL_; ++li) {
    const int box = bi * NL_ + li;
    float s0 = 0.0f, s1 = 0.0f;
    for (int t = tid; t < NT; t += 256) {
      s0 += part[(box * NT + t) * 2 + 0];
      s1 += part[(box * NT + t) * 2 + 1];
    }
    float2 r = block_reduce2(s0, s1, redA, redB, tid);
    if (tid == 0) { Ls[li] = r.x; L1[li] = r.y; }
  }
  __syncthreads();

  if (tid == 0) {
    float beta = c_scalar[0];
    float sumLs = 0.0f, sumL1 = 0.0f, cnt = 0.0f;
    for (int li = 0; li < NL_; ++li) {
      const int box = bi * NL_ + li;
      int x1 = label[box * 5 + 1], y1 = label[box * 5 + 2];
      int x2 = label[box * 5 + 3], y2 = label[box * 5 + 4];
      float v    = (x1 == 0 && y1 == 0 && x2 == 0 && y2 == 0) ? 0.0f : 1.0f;
      float area = (float)((y2 - y1) * (x2 - x1));   // C == 1
      float sa   = fmaxf(area, 1.0f);
      float lsv  = (Ls[li] / sa) * v;
      float l1v  = (L1[li] / sa) * v;
      float ex   = (lsv != 0.0f || l1v != 0.0f) ? 1.0f : 0.0f;
      cnt += ex; sumLs += lsv; sumL1 += l1v;
    }
    float safec = fmaxf(cnt, 1.0f);
    float ls  = (cnt > 0.0f) ? sumLs / safec : 0.0f;
    float lin = (cnt > 0.0f) ? sumL1 / safec : 0.0f;
    out[2 * B_ + bi] = (1.0f - beta) * ls + beta * lin;  // loss_label
    out[5 * B_ + bi] = ls;                               // ls
    out[6 * B_ + bi] = lin;                              // lin
  }
}

// ---------------------------------------------------------------------------
extern "C" void kernel_launch(void* const* d_in, const int* in_sizes, int n_in,
                              void* d_out, int out_size, void* d_ws, size_t ws_size,
                              hipStream_t stream) {
  (void)in_sizes; (void)n_in; (void)out_size; (void)ws_size;
  const float* b_sc  = (const float*)d_in[0];
  const float* c_sc  = (const float*)d_in[1];
  const float* vis   = (const float*)d_in[2];
  // d_in[3] (image_ir) is unused by the reference math.
  const float* gen   = (const float*)d_in[4];
  const int*   label = (const int*)  d_in[5];
  const float* lwth  = (const float*)d_in[6];
  float* out = (float*)d_out;

  float* part1 = (float*)d_ws;                 // [B][NT][2]      = 3200 f
  float* part2 = part1 + (size_t)B_ * NT * 2;  // [B*NL][NT][2]   = 32000 f

  fusion_global_tiles<<<dim3(NTX, NTY, B_),        dim3(32, 8), 0, stream>>>(
      vis, lwth, gen, part1);
  fusion_box_tiles  <<<dim3(NTX, NTY, B_ * NL_),   dim3(32, 8), 0, stream>>>(
      vis, lwth, gen, label, part2);
  finalize_global   <<<dim3(B_), dim3(256), 0, stream>>>(part1, b_sc, out);
  finalize_label    <<<dim3(B_), dim3(256), 0, stream>>>(part2, label, c_sc, out);
}